// ThresholdPruningMGSL_16131897164058
// MI455X (gfx1250) — compile-verified
//
#include <hip/hip_runtime.h>
#include <math.h>

// ---------------------------------------------------------------------------
// Gaussian-HMM forward scan for MI455X (gfx1250).
// Latency-bound sequential chain -> one persistent 8-wave workgroup.
// All O(64^3) math routed through V_WMMA_F32_16X16X4_F32 (f32 for Cholesky
// numerical stability). State resident in LDS (stride-padded).
// ---------------------------------------------------------------------------

typedef __attribute__((ext_vector_type(2))) float v2f;
typedef __attribute__((ext_vector_type(8))) float v8f;

#define D64   64
#define ST    68          // padded LDS row stride (floats) -> conflict-free column reads
#define NTHR  256         // 8 wave32
#define NLAB  50
#define C_LOG2PI 1.8378770664093453f

struct Smem {
  float VAR[D64 * ST];    // carry covariance
  float WORK[D64 * ST];   // vs/lam -> L -> V^-1
  float LINV[D64 * ST];   // L^-1, then reused as product temp / Kt
  float mu[D64], im[D64], ivd[D64], dif[D64];
  float t1[D64], t2[D64], t3[D64], t4[D64];
  float mua[D64], mub[D64];
  float logdet;
  float sc;
};

// ---- 64x64x64 f32 matmul via V_WMMA_F32_16X16X4_F32 -----------------------
// C[m][n] = sum_k opA(A)[m][k] * B[k][n] * (colscale ? colscale[n] : 1)
// 16 tiles of 16x16; 8 waves -> 2 tiles per wave; K in chunks of 4.
template <bool TA>
__device__ __forceinline__ void mm64(const float* __restrict__ A, int lda,
                                     const float* __restrict__ B, int ldb,
                                     float* __restrict__ C, int ldc,
                                     const float* __restrict__ colscale,
                                     int tid) {
  const int wave = tid >> 5;
  const int lane = tid & 31;
  const int half = lane >> 4;   // lanes 16..31 handle K+2 (A/B) and M+8 (C)
  const int lm   = lane & 15;
  for (int t = wave; t < 16; t += 8) {
    const int ti = (t >> 2) << 4;
    const int tj = (t & 3) << 4;
    float cs = 1.0f;
    if (colscale) cs = colscale[tj + lm];
    v8f acc = {};
#pragma unroll
    for (int k0 = 0; k0 < D64; k0 += 4) {
      const int ka = k0 + 2 * half;
      v2f a, b;
      if (TA) {
        a.x = A[ka * lda + ti + lm];
        a.y = A[(ka + 1) * lda + ti + lm];
      } else {
        a.x = A[(ti + lm) * lda + ka];
        a.y = A[(ti + lm) * lda + ka + 1];
      }
      b.x = B[ka * ldb + tj + lm] * cs;
      b.y = B[(ka + 1) * ldb + tj + lm] * cs;
      acc = __builtin_amdgcn_wmma_f32_16x16x4_f32(
          /*neg_a=*/false, a, /*neg_b=*/false, b,
          /*c_mod=*/(short)0, acc, /*reuse_a=*/false, /*reuse_b=*/false);
    }
#pragma unroll
    for (int r = 0; r < 8; ++r)
      C[(ti + r + 8 * half) * ldc + tj + lm] = acc[r];
  }
}

// ---- in-place Cholesky (lower) of 64x64 in LDS; accumulates 2*sum(log Lii)
__device__ void chol64(float* A, float* logdet, int tid) {
  if (tid == 0) *logdet = 0.0f;
  __syncthreads();
  for (int j = 0; j < D64; ++j) {
    if (tid == 0) {
      float d = sqrtf(A[j * ST + j]);
      A[j * ST + j] = d;
      *logdet += 2.0f * logf(d);
    }
    __syncthreads();
    const float dinv = 1.0f / A[j * ST + j];
    for (int i = j + 1 + tid; i < D64; i += NTHR) A[i * ST + j] *= dinv;
    __syncthreads();
    const int rem = D64 - 1 - j;  // full trailing update keeps symmetry
    for (int idx = tid; idx < rem * rem; idx += NTHR) {
      const int i = j + 1 + idx / rem;
      const int k = j + 1 + idx % rem;
      A[i * ST + k] = fmaf(-A[i * ST + j], A[k * ST + j], A[i * ST + k]);
    }
    __syncthreads();
  }
}

// ---- M = L^-1 (lower-tri): 64 independent per-column forward substitutions
__device__ void linv64(const float* L, float* M, int tid) {
  for (int idx = tid; idx < D64 * D64; idx += NTHR)
    M[(idx >> 6) * ST + (idx & 63)] = 0.0f;
  __syncthreads();
  if (tid < D64) {
    const int c = tid;
    for (int i = c; i < D64; ++i) {
      float s = (i == c) ? 1.0f : 0.0f;
      for (int k = c; k < i; ++k) s = fmaf(-L[i * ST + k], M[k * ST + c], s);
      M[i * ST + c] = s / L[i * ST + i];
    }
  }
  __syncthreads();
}

__device__ __forceinline__ void matvec(const float* M, int ld, const float* x,
                                       float* y, int tid) {
  if (tid < D64) {
    float s = 0.0f;
    for (int k = 0; k < D64; ++k) s = fmaf(M[tid * ld + k], x[k], s);
    y[tid] = s;
  }
}
__device__ __forceinline__ void matvecT(const float* M, int ld, const float* x,
                                        float* y, int tid) {
  if (tid < D64) {
    float s = 0.0f;
    for (int k = 0; k < D64; ++k) s = fmaf(M[k * ld + tid], x[k], s);
    y[tid] = s;
  }
}

// ---- gaussian_multi(mu_prev, VAR, im, diag(ivd)) ---------------------------
__device__ void emission(Smem& s, int tid) {
  if (tid < D64) s.dif[tid] = s.mu[tid] - s.im[tid];
  for (int idx = tid; idx < D64 * D64; idx += NTHR) {
    const int i = idx >> 6, j = idx & 63;
    s.WORK[i * ST + j] = s.VAR[i * ST + j] + ((i == j) ? s.ivd[i] : 0.0f);
  }
  __syncthreads();
  chol64(s.WORK, &s.logdet, tid);
  linv64(s.WORK, s.LINV, tid);
  mm64<true>(s.LINV, ST, s.LINV, ST, s.WORK, ST, nullptr, tid);  // WORK = vs^-1
  __syncthreads();
  matvec(s.WORK, ST, s.mu, s.t1, tid);   // t1 = V^-1 mu_prev
  matvec(s.WORK, ST, s.im, s.t2, tid);   // t2 = V^-1 im
  matvec(s.WORK, ST, s.dif, s.t3, tid);  // t3 = V^-1 diff
  __syncthreads();
  if (tid == 0) {
    float q = 0.0f;
    for (int k = 0; k < D64; ++k) q = fmaf(s.dif[k], s.t3[k], q);
    s.sc += -0.5f * (D64 * C_LOG2PI + s.logdet + q);
  }
  matvec(s.VAR, ST, s.t2, s.t4, tid);    // t4 = VAR (V^-1 im)
  __syncthreads();
  // P = VAR @ (V^-1 * diag(ivd))  (column scale fused into B fragment load)
  mm64<false>(s.VAR, ST, s.WORK, ST, s.LINV, ST, s.ivd, tid);
  __syncthreads();
  if (tid < D64) s.mu[tid] = fmaf(s.ivd[tid], s.t1[tid], s.t4[tid]);
  for (int idx = tid; idx < D64 * D64; idx += NTHR) {  // VAR = sym(P)
    const int i = idx >> 6, j = idx & 63;
    s.VAR[i * ST + j] = 0.5f * (s.LINV[i * ST + j] + s.LINV[j * ST + i]);
  }
  __syncthreads();
}

// ---- gaussian_multi_integral_fwd(trans, mu_e, var_e) -----------------------
// tv = 128x128 trans_var in workspace: Saa = tv[i][j], Sab = tv[i][64+j],
// Sbb = tv[64+i][64+j]
__device__ void transition(Smem& s, const float* __restrict__ tv, int tid) {
  if (tid < D64) s.dif[tid] = s.mu[tid] - s.mua[tid];
  for (int idx = tid; idx < D64 * D64; idx += NTHR) {
    const int i = idx >> 6, j = idx & 63;
    s.WORK[i * ST + j] = tv[i * 128 + j] + s.VAR[i * ST + j];  // lam = Saa+var
  }
  __syncthreads();
  chol64(s.WORK, &s.logdet, tid);
  linv64(s.WORK, s.LINV, tid);
  mm64<true>(s.LINV, ST, s.LINV, ST, s.WORK, ST, nullptr, tid);  // WORK = lam^-1
  __syncthreads();
  matvec(s.WORK, ST, s.dif, s.t3, tid);
  __syncthreads();
  if (tid == 0) {
    float q = 0.0f;
    for (int k = 0; k < D64; ++k) q = fmaf(s.dif[k], s.t3[k], q);
    s.sc += -0.5f * (D64 * C_LOG2PI + s.logdet + q);
  }
  // Kt = lam^-1 @ Sab  -> LINV
  mm64<false>(s.WORK, ST, tv + 64, 128, s.LINV, ST, nullptr, tid);
  __syncthreads();
  matvecT(s.LINV, ST, s.dif, s.t4, tid);  // t4 = Kt^T (mu_e - mu_a)
  __syncthreads();
  if (tid < D64) s.mu[tid] = s.mub[tid] + s.t4[tid];
  // Q = Sab^T @ Kt -> WORK (lam^-1 dead)
  mm64<true>(tv + 64, 128, s.LINV, ST, s.WORK, ST, nullptr, tid);
  __syncthreads();
  for (int idx = tid; idx < D64 * D64; idx += NTHR) {  // VAR = sym(Sbb - Q)
    const int i = idx >> 6, j = idx & 63;
    const float aij = tv[(64 + i) * 128 + 64 + j] - s.WORK[i * ST + j];
    const float aji = tv[(64 + j) * 128 + 64 + i] - s.WORK[j * ST + i];
    s.VAR[i * ST + j] = 0.5f * (aij + aji);
  }
  __syncthreads();
}

// ---- prologue: trans_var = trans_cho^T @ trans_cho (128x128, runs once) ----
__global__ void __launch_bounds__(128)
transvar_kernel(const float* __restrict__ tcho, float* __restrict__ tv) {
  const int j = threadIdx.x;   // column
  const int i = blockIdx.y;    // row
  float acc = 0.0f;
  for (int k = 0; k < 128; ++k)
    acc = fmaf(tcho[k * 128 + i], tcho[k * 128 + j], acc);
  tv[i * 128 + j] = acc;
}

// ---- persistent scan kernel ------------------------------------------------
__global__ void __launch_bounds__(NTHR)
hmm_forward_kernel(const int* __restrict__ sentence,
                   const float* __restrict__ mu_emb,
                   const float* __restrict__ cho_emb,
                   const float* __restrict__ trans_mu,
                   const float* __restrict__ tv,
                   const float* __restrict__ out_mu,
                   const float* __restrict__ out_cho,
                   float* __restrict__ out, int T) {
  __shared__ Smem s;
  const int tid = threadIdx.x;

  if (tid < D64) {
    s.mu[tid]  = 0.0f;                 // init_mu = 0
    s.mua[tid] = trans_mu[tid];
    s.mub[tid] = trans_mu[D64 + tid];
  }
  for (int idx = tid; idx < D64 * D64; idx += NTHR) {  // init_var = I
    const int i = idx >> 6, j = idx & 63;
    s.VAR[i * ST + j] = (i == j) ? 1.0f : 0.0f;
  }
  if (tid == 0) s.sc = 0.0f;
  __syncthreads();

  transition(s, tv, tid);  // (s0, m0, v0)

  for (int t = 0; t < T; ++t) {
    const int tok = sentence[t];
    if (tid < D64) {
      s.im[tid] = mu_emb[(long)tok * D64 + tid];
      const float c = cho_emb[(long)tok * D64 + tid];
      s.ivd[tid] = c * c;
    }
    __syncthreads();
    emission(s, tid);
    transition(s, tv, tid);
  }

  // decode: scores[l] = log_gauss(fmu - omu_l, fvar + diag(ocho_l^2)) + fscore
  for (int l = 0; l < NLAB; ++l) {
    if (tid < D64) s.dif[tid] = s.mu[tid] - out_mu[l * D64 + tid];
    for (int idx = tid; idx < D64 * D64; idx += NTHR) {
      const int i = idx >> 6, j = idx & 63;
      float dg = 0.0f;
      if (i == j) { const float c = out_cho[l * D64 + i]; dg = c * c; }
      s.WORK[i * ST + j] = s.VAR[i * ST + j] + dg;
    }
    __syncthreads();
    chol64(s.WORK, &s.logdet, tid);
    if (tid == 0) {
      float q = 0.0f;
      for (int i = 0; i < D64; ++i) {  // z = L^-1 dif; q = z.z
        float v = s.dif[i];
        for (int k = 0; k < i; ++k) v = fmaf(-s.WORK[i * ST + k], s.t3[k], v);
        v /= s.WORK[i * ST + i];
        s.t3[i] = v;
        q = fmaf(v, v, q);
      }
      out[l] = s.sc - 0.5f * (D64 * C_LOG2PI + s.logdet + q);
    }
    __syncthreads();
  }
}

extern "C" void kernel_launch(void* const* d_in, const int* in_sizes, int n_in,
                              void* d_out, int out_size, void* d_ws,
                              size_t ws_size, hipStream_t stream) {
  const int*   sentence  = (const int*)d_in[0];
  // d_in[1] = slen (scalar); sentence length also == in_sizes[0]
  const float* mu_emb    = (const float*)d_in[2];
  const float* cho_emb   = (const float*)d_in[3];
  const float* trans_mu  = (const float*)d_in[4];
  const float* trans_cho = (const float*)d_in[5];
  const float* out_mu    = (const float*)d_in[6];
  const float* out_cho   = (const float*)d_in[7];
  float* out = (float*)d_out;
  float* tv  = (float*)d_ws;  // 128*128 f32 = 64 KB

  const int T = in_sizes[0];

  transvar_kernel<<<dim3(1, 128), dim3(128), 0, stream>>>(trans_cho, tv);
  hmm_forward_kernel<<<1, NTHR, 0, stream>>>(sentence, mu_emb, cho_emb,
                                             trans_mu, tv, out_mu, out_cho,
                                             out, T);
}